// GCN_51634096832829
// MI455X (gfx1250) — compile-verified
//
#include <hip/hip_runtime.h>

// GCN constants from the reference
#define NF 128   // input features
#define NH 128   // hidden
#define NC 40    // classes

typedef float v2f __attribute__((ext_vector_type(2)));
typedef float v8f __attribute__((ext_vector_type(8)));

// ---------------------------------------------------------------------------
// Degree / normalization
// ---------------------------------------------------------------------------
__global__ void deg_init_k(float* __restrict__ deg, int n) {
    int i = blockIdx.x * blockDim.x + threadIdx.x;
    if (i < n) deg[i] = 1.0f;   // self-loop contributes 1 to every node's degree
}

__global__ void deg_accum_k(const long long* __restrict__ dst, float* __restrict__ deg, int e) {
    int i = blockIdx.x * blockDim.x + threadIdx.x;
    if (i < e) unsafeAtomicAdd(&deg[(int)dst[i]], 1.0f);
}

__global__ void dinv_k(const float* __restrict__ deg, float* __restrict__ dinv, int n) {
    int i = blockIdx.x * blockDim.x + threadIdx.x;
    if (i < n) {
        float d = deg[i];
        dinv[i] = (d > 0.0f) ? rsqrtf(d) : 0.0f;   // d >= 1 here, branch kept for fidelity
    }
}

// ---------------------------------------------------------------------------
// fp32 WMMA GEMM:  O[N,Wn] = A[N,128] @ W[128,Wn]
// One wave32 per 16x16 output tile; K looped in steps of 4 via
// V_WMMA_F32_16X16X4_F32 (exact fp32, matches reference numerics).
//
// VGPR layouts (ISA 7.12.2):
//   A 16x4 : lane<16 -> M=lane, K={0,1}; lane>=16 -> M=lane-16, K={2,3}
//            => per-lane contiguous float2 from the A row
//   B 4x16 : VGPR0: lanes0-15 K=0 / lanes16-31 K=2; VGPR1: K=1 / K=3
//   C 16x16: VGPR r: lanes0-15 row r, lanes16-31 row r+8; col = lane&15
// ---------------------------------------------------------------------------
__global__ void __launch_bounds__(256)
gemm_wmma_h128(const float* __restrict__ A, const float* __restrict__ W,
               float* __restrict__ O, int n) {
    const int lane = threadIdx.x & 31;
    const int wv   = threadIdx.x >> 5;          // 0..7 -> column tile of H=128
    const int l15  = lane & 15;
    const int half = lane >> 4;                 // 0: K-pair {0,1}; 1: K-pair {2,3}
    const int row0 = blockIdx.x * 16;
    const int col  = wv * 16 + l15;

    const int arow = min(row0 + l15, n - 1);    // clamp (stores are guarded)
    const float* ap = A + (size_t)arow * NF;

    v8f c = {};
#pragma unroll
    for (int k0 = 0; k0 < NF; k0 += 4) {
        v2f a = *(const v2f*)(ap + k0 + 2 * half);        // 8B aligned (k even)
        v2f b;
        b.x = W[(k0 + 2 * half + 0) * NH + col];
        b.y = W[(k0 + 2 * half + 1) * NH + col];
        c = __builtin_amdgcn_wmma_f32_16x16x4_f32(false, a, false, b,
                                                  (short)0, c, false, false);
    }
    const int rbase = row0 + half * 8;
#pragma unroll
    for (int r = 0; r < 8; ++r)
        if (rbase + r < n) O[(size_t)(rbase + r) * NH + col] = c[r];
}

__global__ void __launch_bounds__(96)
gemm_wmma_h40(const float* __restrict__ A, const float* __restrict__ W,
              float* __restrict__ O, int n) {
    const int lane = threadIdx.x & 31;
    const int wv   = threadIdx.x >> 5;          // 0..2 -> column tiles cover 48 >= 40
    const int l15  = lane & 15;
    const int half = lane >> 4;
    const int row0 = blockIdx.x * 16;
    const int col  = wv * 16 + l15;
    const bool cok = (col < NC);

    const int arow = min(row0 + l15, n - 1);
    const float* ap = A + (size_t)arow * NH;

    v8f c = {};
#pragma unroll
    for (int k0 = 0; k0 < NH; k0 += 4) {
        v2f a = *(const v2f*)(ap + k0 + 2 * half);
        v2f b;
        b.x = cok ? W[(k0 + 2 * half + 0) * NC + col] : 0.0f;
        b.y = cok ? W[(k0 + 2 * half + 1) * NC + col] : 0.0f;
        c = __builtin_amdgcn_wmma_f32_16x16x4_f32(false, a, false, b,
                                                  (short)0, c, false, false);
    }
    const int rbase = row0 + half * 8;
#pragma unroll
    for (int r = 0; r < 8; ++r)
        if (cok && (rbase + r < n)) O[(size_t)(rbase + r) * NC + col] = c[r];
}

// ---------------------------------------------------------------------------
// Self-loop term + bias initializers (also fully initialize the accumulators
// before the edge-atomic kernels run; harness poisons d_out/d_ws with 0xAA).
// ---------------------------------------------------------------------------
__global__ void init_acc128_k(const float* __restrict__ hm, const float* __restrict__ dinv,
                              const float* __restrict__ bias, float* __restrict__ acc, int n) {
    int i = blockIdx.x * blockDim.x + threadIdx.x;
    if (i < n * NH) {
        int node = i >> 7, f = i & (NH - 1);
        float di = dinv[node];
        acc[i] = hm[i] * di * di + bias[f];
    }
}

__global__ void init_acc40_k(const float* __restrict__ hm, const float* __restrict__ dinv,
                             const float* __restrict__ bias, float* __restrict__ acc, int n) {
    int i = blockIdx.x * blockDim.x + threadIdx.x;
    if (i < n * NC) {
        int node = i / NC, f = i % NC;
        float di = dinv[node];
        acc[i] = hm[i] * di * di + bias[f];
    }
}

__global__ void relu_k(float* __restrict__ h, int total) {
    int i = blockIdx.x * blockDim.x + threadIdx.x;
    if (i < total) h[i] = fmaxf(h[i], 0.0f);
}

// ---------------------------------------------------------------------------
// Edge scatter: one wave32 per edge. src/dst/norm are wave-uniform (scalarize),
// message row read as float4 per lane, hardware f32 atomics into the dest row.
// Traffic is L2-resident (whole working set << 192 MB L2).
// ---------------------------------------------------------------------------
__global__ void __launch_bounds__(256)
edge_scatter128_k(const long long* __restrict__ src, const long long* __restrict__ dst,
                  const float* __restrict__ dinv, const float* __restrict__ hm,
                  float* __restrict__ acc, int e) {
    int wave = (int)((blockIdx.x * (unsigned)blockDim.x + threadIdx.x) >> 5);
    int lane = threadIdx.x & 31;
    if (wave >= e) return;
    int s = (int)src[wave];
    int d = (int)dst[wave];
    float nrm = dinv[s] * dinv[d];
    float4 v = ((const float4*)(hm + (size_t)s * NH))[lane];   // 128 = 32 lanes x 4
    float* o = acc + (size_t)d * NH + lane * 4;
    unsafeAtomicAdd(o + 0, v.x * nrm);
    unsafeAtomicAdd(o + 1, v.y * nrm);
    unsafeAtomicAdd(o + 2, v.z * nrm);
    unsafeAtomicAdd(o + 3, v.w * nrm);
}

__global__ void __launch_bounds__(256)
edge_scatter40_k(const long long* __restrict__ src, const long long* __restrict__ dst,
                 const float* __restrict__ dinv, const float* __restrict__ hm,
                 float* __restrict__ acc, int e) {
    int wave = (int)((blockIdx.x * (unsigned)blockDim.x + threadIdx.x) >> 5);
    int lane = threadIdx.x & 31;
    if (wave >= e) return;
    int s = (int)src[wave];
    int d = (int)dst[wave];
    float nrm = dinv[s] * dinv[d];
    const float* m = hm + (size_t)s * NC;
    float* o = acc + (size_t)d * NC;
    unsafeAtomicAdd(o + lane, m[lane] * nrm);                  // features 0..31
    if (lane < NC - 32)
        unsafeAtomicAdd(o + 32 + lane, m[32 + lane] * nrm);    // features 32..39
}

// ---------------------------------------------------------------------------
// Host-side orchestration (all launches on `stream`, graph-capture safe)
// ---------------------------------------------------------------------------
extern "C" void kernel_launch(void* const* d_in, const int* in_sizes, int n_in,
                              void* d_out, int out_size, void* d_ws, size_t ws_size,
                              hipStream_t stream) {
    const float*     x   = (const float*)d_in[0];
    const long long* ei  = (const long long*)d_in[1];   // int64 [2, E]
    const float*     W1  = (const float*)d_in[2];
    const float*     b1  = (const float*)d_in[3];
    const float*     W2  = (const float*)d_in[4];
    const float*     b2  = (const float*)d_in[5];
    float*           out = (float*)d_out;

    const int n = in_sizes[0] / NF;          // 10000
    const int e = in_sizes[1] / 2;           // 640000
    const long long* srcIdx = ei;
    const long long* dstIdx = ei + e;

    // Workspace layout (floats): deg[n] | dinv[n] | hm1[n*128] | h1[n*128] | hm2[n*40]
    float* deg  = (float*)d_ws;
    float* dinv = deg  + n;
    float* hm1  = dinv + n;
    float* h1   = hm1  + (size_t)n * NH;
    float* hm2  = h1   + (size_t)n * NH;

    const int mt = (n + 15) / 16;            // 625 row tiles
    const int eb = (e + 7) / 8;              // edge blocks (8 waves/block)

    // 1) degree + normalization
    deg_init_k <<<(n + 255) / 256, 256, 0, stream>>>(deg, n);
    deg_accum_k<<<(e + 255) / 256, 256, 0, stream>>>(dstIdx, deg, e);
    dinv_k     <<<(n + 255) / 256, 256, 0, stream>>>(deg, dinv, n);

    // 2) layer 1: transform -> self-loop+bias init -> edge scatter -> ReLU
    gemm_wmma_h128   <<<mt, 256, 0, stream>>>(x, W1, hm1, n);
    init_acc128_k    <<<(n * NH + 255) / 256, 256, 0, stream>>>(hm1, dinv, b1, h1, n);
    edge_scatter128_k<<<eb, 256, 0, stream>>>(srcIdx, dstIdx, dinv, hm1, h1, e);
    relu_k           <<<(n * NH + 255) / 256, 256, 0, stream>>>(h1, n * NH);

    // 3) layer 2: transform -> self-loop+bias init -> edge scatter
    gemm_wmma_h40   <<<mt, 96, 0, stream>>>(h1, W2, hm2, n);
    init_acc40_k    <<<(n * NC + 255) / 256, 256, 0, stream>>>(hm2, dinv, b2, out, n);
    edge_scatter40_k<<<eb, 256, 0, stream>>>(srcIdx, dstIdx, dinv, hm2, out, e);
}